// HashGridEncoder2D_62122406969915
// MI455X (gfx1250) — compile-verified
//
#include <hip/hip_runtime.h>
#include <math.h>
#include <stdint.h>

// ---------------------------------------------------------------------------
// HashGridEncoder2D for gfx1250 (MI455X)
//  - 64M random 8B gathers from a 22.1MB table (L2-resident on 192MB L2)
//  - levels 0..5 (109KB) staged into LDS via CDNA5 async global->LDS copies
//    (global_load_async_to_lds_b128 + s_wait_asynccnt), 2 workgroups/WGP
//  - hashed-level mask specialized at compile time (0xF000 fast path)
//  - packed V_PK_FMA_F32 bilinear interpolation (F=2 features per op)
//  - per-pair NT stores to cap live VGPRs (occupancy hides gather latency)
// ---------------------------------------------------------------------------

#define TBL_T      524288
#define TBL_MASK   (TBL_T - 1)
#define NLEV       16
#define LDS_LEVELS 6          // levels [0,6) read from LDS (13670 entries = 109360 B)
#define EXP_MASK   0xF000u    // expected hashed-level mask (levels 12..15)

typedef float vf2 __attribute__((ext_vector_type(2)));
typedef float vf4 __attribute__((ext_vector_type(4)));

struct GridMeta {
  int      res[NLEV];
  int      off[NLEV];
  unsigned hashed_mask;
  int      prime;
  int      lds_floats;  // floats staged into LDS == off[LDS_LEVELS]*2
  unsigned npts;
};

#if defined(__has_builtin)
#  if __has_builtin(__builtin_amdgcn_global_load_async_to_lds_b128)
#    define HAVE_ASYNC_LDS 1
#  endif
#  if __has_builtin(__builtin_amdgcn_s_wait_asynccnt)
#    define HAVE_WAIT_ASYNC 1
#  endif
#endif
#ifndef HAVE_ASYNC_LDS
#  define HAVE_ASYNC_LDS 0
#endif
#ifndef HAVE_WAIT_ASYNC
#  define HAVE_WAIT_ASYNC 0
#endif

#if HAVE_ASYNC_LDS
// Exact parameter pointee type per hipcc diagnostic:
//   "__attribute__((__vector_size__(4 * sizeof(int)))) int __device__ *"
typedef int b128_t __attribute__((vector_size(4 * sizeof(int))));
typedef __attribute__((address_space(1))) b128_t gb128_t;   // global (__device__)
typedef __attribute__((address_space(3))) b128_t lb128_t;   // LDS (__local)

__device__ __forceinline__ void async_cp16(const float* g, float* l) {
  // AS1 ptr value == flat value; AS3 value == low 32 bits of flat LDS ptr
  // (flat LDS aperture: addr[31:0] is the LDS offset, per CDNA5 ISA 10.2).
  __builtin_amdgcn_global_load_async_to_lds_b128(
      (gb128_t*)(uintptr_t)g,
      (lb128_t*)(uint32_t)(uintptr_t)l,
      0, 0);
}
#endif

__device__ __forceinline__ void wait_async_done() {
#if HAVE_ASYNC_LDS
#  if HAVE_WAIT_ASYNC
  __builtin_amdgcn_s_wait_asynccnt(0);
#  else
  asm volatile("s_wait_asynccnt 0" ::: "memory");
#  endif
#endif
}

// One level of the encoder: 4 gathers + packed bilinear blend.
template <bool SPEC, unsigned HMASK>
__device__ __forceinline__ vf2 encode_level(int l, vf2 uvp, const GridMeta& m,
                                            const float* __restrict__ lat,
                                            const float* __restrict__ lds_lat) {
  const int   res  = m.res[l];
  const int   lofs = m.off[l];
  const float fr   = (float)res;

  const float sx = uvp.x * fr;
  const float sy = uvp.y * fr;
  const float fx = floorf(sx);
  const float fy = floorf(sy);
  const int   x0 = (int)fx;
  const int   y0 = (int)fy;
  const float px = sx - fx;        // in [0,1): clip in reference is a no-op
  const float py = sy - fy;
  const float qx = 1.0f - px;
  const float qy = 1.0f - py;
  const float w00 = qx * qy;       // corner (x0,   y0  )
  const float w01 = qx * py;       // corner (x0,   y0+1)
  const float w10 = px * qy;       // corner (x0+1, y0  )
  const float w11 = px * py;       // corner (x0+1, y0+1)

  const bool hashed =
      SPEC ? (((HMASK >> l) & 1u) != 0u) : ((m.hashed_mask & (1u << l)) != 0u);

  int i00, i01, i10, i11;
  if (hashed) {                    // compile-time when SPEC (l is constant)
    const int hy0 = y0 * m.prime;
    const int hy1 = hy0 + m.prime;
    i00 = (x0 ^ hy0) & TBL_MASK;
    i01 = (x0 ^ hy1) & TBL_MASK;
    i10 = ((x0 + 1) ^ hy0) & TBL_MASK;
    i11 = ((x0 + 1) ^ hy1) & TBL_MASK;
  } else {
    const int base = x0 * res + y0;
    i00 = base;
    i01 = base + 1;                // folds into 8B imm load offset
    i10 = base + res;
    i11 = base + res + 1;
  }

  vf2 c00, c01, c10, c11;
  if (l < LDS_LEVELS) {            // compile-time: ds_load_b64 gathers
    const vf2* p = (const vf2*)lds_lat;
    c00 = p[lofs + i00];
    c01 = p[lofs + i01];
    c10 = p[lofs + i10];
    c11 = p[lofs + i11];
  } else {                         // global_load_b64 gathers (L2-hot)
    const vf2* p = (const vf2*)lat;
    c00 = p[lofs + i00];
    c01 = p[lofs + i01];
    c10 = p[lofs + i10];
    c11 = p[lofs + i11];
  }

  // packed (2-feature) bilinear blend -> V_PK_FMA_F32 per term
  return c00 * w00 + c01 * w01 + c10 * w10 + c11 * w11;
}

template <bool SPEC, unsigned HMASK>
__global__ __launch_bounds__(1024) void hashgrid2d_kernel(
    const float* __restrict__ uv,
    const float* __restrict__ lat,
    float* __restrict__ out,
    GridMeta m)
{
  extern __shared__ float lds_lat[];

  // ---- stage levels [0, LDS_LEVELS) of the latent table into LDS ----------
  {
    const int nchunk = m.lds_floats >> 2;  // 16B chunks
    for (int c = threadIdx.x; c < nchunk; c += blockDim.x) {
#if HAVE_ASYNC_LDS
      async_cp16(lat + 4 * c, lds_lat + 4 * c);
#else
      ((vf4*)lds_lat)[c] = ((const vf4*)lat)[c];
#endif
    }
    for (int c = (nchunk << 2) + (int)threadIdx.x; c < m.lds_floats;
         c += blockDim.x)
      lds_lat[c] = lat[c];  // tail (empty for this config)
    wait_async_done();
    __syncthreads();
  }

  const unsigned gid = blockIdx.x * blockDim.x + threadIdx.x;
  if (gid >= m.npts) return;

  const vf2 uvp = __builtin_nontemporal_load(((const vf2*)uv) + gid);

  // 128 contiguous bytes per point; non-temporal so the 134MB output stream
  // does not evict the 22MB latent table from L2. Store per level-pair to
  // keep live VGPRs low (occupancy is what hides random-gather latency).
  vf4* op = (vf4*)(out + (size_t)gid * (2 * NLEV));
#pragma unroll
  for (int k = 0; k < NLEV / 2; ++k) {
    const vf2 a = encode_level<SPEC, HMASK>(2 * k + 0, uvp, m, lat, lds_lat);
    const vf2 b = encode_level<SPEC, HMASK>(2 * k + 1, uvp, m, lat, lds_lat);
    vf4 v;
    v.x = a.x;
    v.y = a.y;
    v.z = b.x;
    v.w = b.y;
    __builtin_nontemporal_store(v, op + k);
  }
}

// ---------------------------------------------------------------------------

static bool is_prime_host(int v) {
  if (v < 2) return false;
  for (int p = 2; (long)p * p <= (long)v; ++p)
    if (v % p == 0) return false;
  return true;
}

extern "C" void kernel_launch(void* const* d_in, const int* in_sizes, int n_in,
                              void* d_out, int out_size, void* d_ws, size_t ws_size,
                              hipStream_t stream) {
  (void)n_in; (void)out_size; (void)d_ws; (void)ws_size;

  const float* uv  = (const float*)d_in[0];
  const float* lat = (const float*)d_in[1];
  float*       out = (float*)d_out;

  // Replicate the Python metadata with the same libm double math so the
  // int() truncations (e.g. 153.99 -> 153) match exactly.
  GridMeta m;
  const double b = exp((log(2048.0) - log(16.0)) / 15.0);
  int offs = 0;
  m.hashed_mask = 0u;
  for (int i = 0; i < NLEV; ++i) {
    const int res = (int)(16.0 * pow(b, (double)i));
    long n = (long)(res + 1) * (long)(res + 1);
    m.res[i] = res;
    m.off[i] = offs;
    if (n > (long)TBL_T) { m.hashed_mask |= (1u << i); n = TBL_T; }
    offs += (int)n;
  }
  int pr = 1 << 17;
  while (!is_prime_host(pr)) ++pr;   // -> 131101
  m.prime      = pr;
  m.npts       = (unsigned)(in_sizes[0] / 2);
  m.lds_floats = m.off[LDS_LEVELS] * 2;           // 13670 entries -> 27340 floats

  const size_t shbytes = (size_t)m.lds_floats * sizeof(float);  // 109360 B
  const dim3 block(1024);
  const dim3 grid((m.npts + 1023u) / 1024u);

  if (m.hashed_mask == EXP_MASK) {
    hashgrid2d_kernel<true, EXP_MASK><<<grid, block, shbytes, stream>>>(uv, lat, out, m);
  } else {
    hashgrid2d_kernel<false, 0u><<<grid, block, shbytes, stream>>>(uv, lat, out, m);
  }
}